// GATEDecoderLayer_75084618268884
// MI455X (gfx1250) — compile-verified
//
#include <hip/hip_runtime.h>
#include <hip/hip_bf16.h>

// ---------------------------------------------------------------------------
// GAT-style layer:
//   hp  = h @ W_T                          [N,64] x [64,64] -> [N,64]   (WMMA)
//   msg = hp[col] * attn[:,None]           gather + scale
//   out = segment_sum(msg, row)            scatter-add (L2-resident atomics)
// N = 50000 (divisible by 16), E = 1.6M, IN=OUT=64.
// ---------------------------------------------------------------------------

typedef __attribute__((ext_vector_type(2))) float v2f;
typedef __attribute__((ext_vector_type(8))) float v8f;

#define IN_DIM  64
#define OUT_DIM 64

// ---- Kernel 1: hp = h @ W_T via V_WMMA_F32_16X16X4_F32 ---------------------
// blockDim = 128 (4 waves). Wave w computes the 16x16 tile at
// rows [blockIdx.x*16, +16), cols [w*16, +16). K loop: 16 x (K=4) WMMAs.
__global__ void __launch_bounds__(128)
gat_gemm_wmma(const float* __restrict__ h, const float* __restrict__ Wt,
              float* __restrict__ hp) {
    const int wave    = threadIdx.x >> 5;    // 0..3  -> column tile
    const int lane    = threadIdx.x & 31;
    const int half    = lane >> 4;           // 0: lanes 0-15, 1: lanes 16-31
    const int m       = lane & 15;
    const int rowBase = blockIdx.x * 16;
    const int colBase = wave * 16;

    const float* __restrict__ arow = h + (rowBase + m) * IN_DIM;

    v8f acc = {};
#pragma unroll
    for (int k0 = 0; k0 < IN_DIM; k0 += 4) {
        const int k = k0 + 2 * half;         // this half-wave's K pair base
        v2f a, b;
        // A 16x4 f32: VGPR0 = K={0|2}, VGPR1 = K={1|3} (per half-wave)
        a.x = arow[k];
        a.y = arow[k + 1];
        // B 4x16 f32: row K striped across lanes within a VGPR, same half split
        b.x = Wt[k * OUT_DIM + colBase + m];
        b.y = Wt[(k + 1) * OUT_DIM + colBase + m];
        acc = __builtin_amdgcn_wmma_f32_16x16x4_f32(
            /*neg_a=*/false, a, /*neg_b=*/false, b,
            /*c_mod=*/(short)0, acc, /*reuse_a=*/false, /*reuse_b=*/false);
    }

    // D 16x16 f32: VGPR r -> M = r + 8*half, N = m
#pragma unroll
    for (int r = 0; r < 8; ++r) {
        hp[(rowBase + r + 8 * half) * OUT_DIM + colBase + m] = acc[r];
    }
}

// ---- Kernel 2: zero the output (required every launch) ---------------------
__global__ void gat_zero(float4* __restrict__ out, int n4) {
    int i = blockIdx.x * blockDim.x + threadIdx.x;
    if (i < n4) out[i] = make_float4(0.f, 0.f, 0.f, 0.f);
}

// ---- Kernel 3: gather + scale + scatter-add --------------------------------
// One wave per edge. Lane handles 2 consecutive floats: coalesced float2 load
// of hp (L2-resident), two non-returning global_atomic_add_f32 into out.
__global__ void __launch_bounds__(256)
gat_scatter(const float* __restrict__ hp, const int* __restrict__ row,
            const int* __restrict__ col, const float* __restrict__ attn,
            float* __restrict__ out, int n_edges) {
    const int idx  = blockIdx.x * blockDim.x + threadIdx.x;
    const int e    = idx >> 5;
    const int lane = idx & 31;
    if (e >= n_edges) return;

    const int   r = row[e];                  // wave-uniform scalar loads
    const int   c = col[e];
    const float a = attn[e];

    const float2 v = *reinterpret_cast<const float2*>(hp + c * OUT_DIM + lane * 2);
    float* dst = out + (long long)r * OUT_DIM + lane * 2;
    unsafeAtomicAdd(dst,     v.x * a);       // global_atomic_add_f32 (no rtn)
    unsafeAtomicAdd(dst + 1, v.y * a);
}

// ---------------------------------------------------------------------------
extern "C" void kernel_launch(void* const* d_in, const int* in_sizes, int n_in,
                              void* d_out, int out_size, void* d_ws, size_t ws_size,
                              hipStream_t stream) {
    const float* h    = (const float*)d_in[0];   // [N, 64]
    const int*   eidx = (const int*)d_in[1];     // [2, E] flat
    const float* attn = (const float*)d_in[2];   // [E]
    const float* Wt   = (const float*)d_in[3];   // [64, 64]

    const int N = in_sizes[0] / IN_DIM;          // 50000
    const int E = in_sizes[2];                   // 1600000
    const int* row = eidx;                       // edge_index[0]
    const int* col = eidx + E;                   // edge_index[1]

    float* out = (float*)d_out;                  // [N, 64]
    float* hp  = (float*)d_ws;                   // [N, 64] scratch

    // 1) projection GEMM (WMMA): one block = 16 rows x all 64 cols
    gat_gemm_wmma<<<N / 16, 128, 0, stream>>>(h, Wt, hp);

    // 2) zero out (N*64 floats, float4-wide)
    const int n4 = (N * OUT_DIM) / 4;
    gat_zero<<<(n4 + 255) / 256, 256, 0, stream>>>((float4*)out, n4);

    // 3) scatter-add: 32 lanes per edge
    const long long threads = (long long)E * 32;
    const int blocks = (int)((threads + 255) / 256);
    gat_scatter<<<blocks, 256, 0, stream>>>(hp, row, col, attn, out, E);
}